// GraphCNN_86964497809599
// MI455X (gfx1250) — compile-verified
//
#include <hip/hip_runtime.h>
#include <hip/hip_bf16.h>

#define N_NODES 50000
#define N_EDGES 800000
#define N_GRAPHS 128
#define N_LAYERS 5
#define IN_DIM 200
#define HID 128
#define BN_EPS 1e-5f

typedef float v2f __attribute__((ext_vector_type(2)));
typedef float v8f __attribute__((ext_vector_type(8)));

// ------- scatter-add over edges (float4 gather + 4 f32 atomics) -------------
template <int D>
__global__ void scatter_add_k(const float* __restrict__ h,
                              const int* __restrict__ row,
                              const int* __restrict__ col,
                              float* __restrict__ pooled, int totalq) {
  constexpr int Q = D / 4;
  int i = blockIdx.x * blockDim.x + threadIdx.x;
  if (i >= totalq) return;
  int e = i / Q;                       // Q is constexpr -> mul/shift
  int q = i - e * Q;
  const float4 v = ((const float4*)(h + (long)col[e] * D))[q];
  float* dst = pooled + (long)row[e] * D + q * 4;
  atomicAdd(dst + 0, v.x);
  atomicAdd(dst + 1, v.y);
  atomicAdd(dst + 2, v.z);
  atomicAdd(dst + 3, v.w);
}

// ------- WMMA fp32 GEMM, 80 rows x 128 cols per block ------------------------
// MODE 0: A-stage computes  A + (1+eps[l])*selfh   (GIN self term fused)
// MODE 1: A-stage computes  relu(A*scale[c]+shift[c])  (BN1+ReLU fused)
template <int K, int MODE, int MT>
__global__ void gemm_wmma_k(const float* __restrict__ A, const float* __restrict__ W,
                            const float* __restrict__ bias, float* __restrict__ out,
                            const float* __restrict__ selfh, const float* __restrict__ eps,
                            int l, const float* __restrict__ scale,
                            const float* __restrict__ shift) {
  constexpr int ROWS = 16 * MT;
  constexpr int LDA  = K + 4;          // conflict-free for both lane halves
  __shared__ float As[ROWS * LDA];
  const int tid  = threadIdx.x;
  const int row0 = blockIdx.x * ROWS;

  float selfs = 0.f;
  if (MODE == 0) selfs = 1.0f + eps[l];

  for (int idx = tid; idx < ROWS * K; idx += 256) {
    int r = idx / K, c = idx - r * K;
    long gi = (long)(row0 + r) * K + c;
    float v = A[gi];
    if (MODE == 0) {
      v += selfs * selfh[gi];
    } else {
      v = v * scale[c] + shift[c];
      v = v > 0.f ? v : 0.f;
    }
    As[r * LDA + c] = v;
  }
  __syncthreads();

  const int wave = tid >> 5;
  const int lane = tid & 31;
  const int half = lane >> 4;          // 0: lanes 0-15, 1: lanes 16-31
  const int mn   = lane & 15;
  const int kb   = half * 2;           // K sub-offset within fragment
  const int ncol = wave * 16 + mn;

  v8f acc[MT] = {};
#pragma unroll 4
  for (int k = 0; k < K; k += 4) {
    v2f b;                             // B 4x16 frag, reused across MT tiles
    b.x = W[(k + kb) * HID + ncol];
    b.y = W[(k + kb + 1) * HID + ncol];
#pragma unroll
    for (int t = 0; t < MT; ++t) {
      v2f a;                           // A 16x4 frag from LDS
      a.x = As[(t * 16 + mn) * LDA + k + kb];
      a.y = As[(t * 16 + mn) * LDA + k + kb + 1];
      acc[t] = __builtin_amdgcn_wmma_f32_16x16x4_f32(false, a, false, b,
                                                     (short)0, acc[t], false, false);
    }
  }

  const float bv = bias[ncol];
#pragma unroll
  for (int t = 0; t < MT; ++t) {
#pragma unroll
    for (int i = 0; i < 8; ++i) {
      int m = t * 16 + i + 8 * half;   // C/D layout: vgpr i -> M=i / i+8
      out[(long)(row0 + m) * HID + ncol] = acc[t][i] + bv;
    }
  }
}

// ------- per-channel batch stats (sum, sumsq) --------------------------------
__global__ void bn_stats_k(const float* __restrict__ z, float* __restrict__ sums,
                           float* __restrict__ sqs, int nrows) {
  int c  = threadIdx.x & 127;
  int rb = threadIdx.x >> 7;
  float s = 0.f, q = 0.f;
  for (int r = blockIdx.x * 2 + rb; r < nrows; r += gridDim.x * 2) {
    float v = z[(long)r * HID + c];
    s += v; q += v * v;
  }
  atomicAdd(&sums[c], s);
  atomicAdd(&sqs[c], q);
}

__global__ void bn_finalize_k(const float* __restrict__ sums, const float* __restrict__ sqs,
                              const float* __restrict__ g, const float* __restrict__ b,
                              float* __restrict__ scale, float* __restrict__ shift,
                              float inv_n) {
  int c = threadIdx.x;
  float mean = sums[c] * inv_n;
  float var  = sqs[c] * inv_n - mean * mean;
  float s    = g[c] * rsqrtf(var + BN_EPS);
  scale[c] = s;
  shift[c] = b[c] - mean * s;
}

// ------- BN2 affine + ReLU + write h + fused graph pooling -------------------
__global__ void bn_relu_pool_k(const float* __restrict__ z, const float* __restrict__ scale,
                               const float* __restrict__ shift, const int* __restrict__ batch,
                               float* __restrict__ hout, float* __restrict__ pool, int n) {
  int i = blockIdx.x * blockDim.x + threadIdx.x;
  if (i >= n) return;
  int c    = i & 127;
  int node = i >> 7;
  float v = z[i] * scale[c] + shift[c];
  v = v > 0.f ? v : 0.f;
  hout[i] = v;
  if (v != 0.f) atomicAdd(&pool[(long)batch[node] * HID + c], v);
}

// ------- graph pooling of raw x ----------------------------------------------
template <int D>
__global__ void graph_pool_k(const float* __restrict__ h, const int* __restrict__ batch,
                             float* __restrict__ out, int total) {
  int i = blockIdx.x * blockDim.x + threadIdx.x;
  if (i >= total) return;
  int node = i / D;
  int d    = i - node * D;
  atomicAdd(&out[(long)batch[node] * D + d], h[i]);
}

extern "C" void kernel_launch(void* const* d_in, const int* in_sizes, int n_in,
                              void* d_out, int out_size, void* d_ws, size_t ws_size,
                              hipStream_t stream) {
  const float* x     = (const float*)d_in[0];
  const int*   ei    = (const int*)d_in[1];
  const int*   rowp  = ei;
  const int*   colp  = ei + N_EDGES;
  const int*   batch = (const int*)d_in[2];
  const float* eps   = (const float*)d_in[3];
  const float* w1_0  = (const float*)d_in[4];
  const float* w1    = (const float*)d_in[5];
  const float* b1    = (const float*)d_in[6];
  const float* w2    = (const float*)d_in[7];
  const float* b2    = (const float*)d_in[8];
  const float* bn1g  = (const float*)d_in[9];
  const float* bn1b  = (const float*)d_in[10];
  const float* bn2g  = (const float*)d_in[11];
  const float* bn2b  = (const float*)d_in[12];
  float* out = (float*)d_out;

  // workspace layout (floats)
  float* pooled = (float*)d_ws;                         // 50000*200 max
  float* zbuf   = pooled + (size_t)N_NODES * IN_DIM;    // 50000*128
  float* hbuf   = zbuf   + (size_t)N_NODES * HID;       // 50000*128
  float* sums   = hbuf   + (size_t)N_NODES * HID;       // 128
  float* sqs    = sums + 128;
  float* scale  = sqs  + 128;
  float* shift  = scale + 128;

  const int poolHdr = N_GRAPHS * IN_DIM + N_LAYERS * N_GRAPHS * HID; // 107520
  hipMemsetAsync(out, 0, (size_t)poolHdr * sizeof(float), stream);

  { // pooled_layers[0]: pool raw x (50000x200)
    int total = N_NODES * IN_DIM;
    graph_pool_k<IN_DIM><<<(total + 255) / 256, 256, 0, stream>>>(x, batch, out, total);
  }

  const float* h = x;
  const int t128 = N_NODES * HID;
  constexpr int MT = 5;                 // 3125 tiles = 625 blocks * 5
  const int gemmGrid = (N_NODES / 16) / MT;

  for (int l = 0; l < N_LAYERS; ++l) {
    const int D = (l == 0) ? IN_DIM : HID;
    hipMemsetAsync(pooled, 0, (size_t)N_NODES * D * sizeof(float), stream);

    int totalq = N_EDGES * (D / 4);
    if (l == 0)
      scatter_add_k<IN_DIM><<<(totalq + 255) / 256, 256, 0, stream>>>(h, rowp, colp, pooled, totalq);
    else
      scatter_add_k<HID><<<(totalq + 255) / 256, 256, 0, stream>>>(h, rowp, colp, pooled, totalq);

    // ---- GEMM1 (self-term fused): z1 = (pooled + (1+eps)h) @ W1 + b1 ----
    const float* W1 = (l == 0) ? w1_0 : (w1 + (size_t)(l - 1) * HID * HID);
    if (l == 0)
      gemm_wmma_k<IN_DIM, 0, MT><<<gemmGrid, 256, 0, stream>>>(
          pooled, W1, b1 + l * HID, zbuf, h, eps, l, nullptr, nullptr);
    else
      gemm_wmma_k<HID, 0, MT><<<gemmGrid, 256, 0, stream>>>(
          pooled, W1, b1 + l * HID, zbuf, h, eps, l, nullptr, nullptr);

    // ---- BN1 stats ----
    hipMemsetAsync(sums, 0, 2 * 128 * sizeof(float), stream);
    bn_stats_k<<<256, 256, 0, stream>>>(zbuf, sums, sqs, N_NODES);
    bn_finalize_k<<<1, 128, 0, stream>>>(sums, sqs, bn1g + l * HID, bn1b + l * HID,
                                         scale, shift, 1.0f / N_NODES);

    // ---- GEMM2 (BN1 affine+ReLU fused into A-stage): z2 -> reuse `pooled` ----
    gemm_wmma_k<HID, 1, MT><<<gemmGrid, 256, 0, stream>>>(
        zbuf, w2 + (size_t)l * HID * HID, b2 + l * HID, pooled,
        nullptr, nullptr, l, scale, shift);

    // ---- BN2 stats ----
    hipMemsetAsync(sums, 0, 2 * 128 * sizeof(float), stream);
    bn_stats_k<<<256, 256, 0, stream>>>(pooled, sums, sqs, N_NODES);
    bn_finalize_k<<<1, 128, 0, stream>>>(sums, sqs, bn2g + l * HID, bn2b + l * HID,
                                         scale, shift, 1.0f / N_NODES);

    // ---- BN2 apply + ReLU + write h + fused graph pooling ----
    float* hdst = (l == N_LAYERS - 1) ? (out + poolHdr) : hbuf;
    bn_relu_pool_k<<<(t128 + 255) / 256, 256, 0, stream>>>(
        pooled, scale, shift, batch, hdst,
        out + N_GRAPHS * IN_DIM + l * N_GRAPHS * HID, t128);

    h = hdst;
  }
}